// LAdapter_7516192768195
// MI455X (gfx1250) — compile-verified
//
#include <hip/hip_runtime.h>

#define L_CTX 77
#define N_NODES 4500
#define N_PAD 4512          // 282 * 16
#define DIM 768
#define M_TILES 282         // N_PAD / 16
#define N_TILES 48          // DIM / 16
#define WAVES_PER_BLOCK 8

typedef __attribute__((ext_vector_type(2))) float v2f;
typedef __attribute__((ext_vector_type(8))) float v8f;

// ---------------------------------------------------------------------------
// 1) Stream-copy embeddings -> d_out (dominant cost: ~2.1 GB @ 23.3 TB/s)
// ---------------------------------------------------------------------------
__global__ void copy_emb_kernel(const float4* __restrict__ src,
                                float4* __restrict__ dst, long long n4) {
    long long i = (long long)blockIdx.x * blockDim.x + threadIdx.x;
    if (i < n4) dst[i] = src[i];
}

// ---------------------------------------------------------------------------
// 2) Gather EOT rows into padded [N_PAD, DIM] buffer; zero pad rows.
//    Also bit-copies eot_idx into the tail of d_out (second tuple output).
// ---------------------------------------------------------------------------
__global__ void gather_eot_kernel(const float* __restrict__ emb,
                                  const int* __restrict__ eot_idx,
                                  float* __restrict__ x0,
                                  int* __restrict__ out_idx_tail) {
    long long i = (long long)blockIdx.x * blockDim.x + threadIdx.x;
    long long total = (long long)N_PAD * DIM;
    if (i >= total) return;
    int b = (int)(i / DIM);
    int d = (int)(i % DIM);
    if (b < N_NODES) {
        int t = eot_idx[b];
        x0[i] = emb[((long long)t * N_NODES + b) * DIM + d];
        if (d == 0) out_idx_tail[b] = t;   // passthrough output
    } else {
        x0[i] = 0.0f;                      // pad rows
    }
}

// ---------------------------------------------------------------------------
// Zero a padded [N_PAD, DIM] buffer (SpMM accumulator init)
// ---------------------------------------------------------------------------
__global__ void zero_kernel(float4* __restrict__ dst, long long n4) {
    long long i = (long long)blockIdx.x * blockDim.x + threadIdx.x;
    if (i < n4) dst[i] = make_float4(0.f, 0.f, 0.f, 0.f);
}

// ---------------------------------------------------------------------------
// 3) SpMM: dst[row] += w * src[col], one wave per edge, lanes stride DIM.
//    global_atomic_add_f32 path via unsafeAtomicAdd.
// ---------------------------------------------------------------------------
__global__ void spmm_kernel(const int* __restrict__ rows,
                            const int* __restrict__ cols,
                            const float* __restrict__ w,
                            const float* __restrict__ src,
                            float* __restrict__ dst, int n_edges) {
    int wave = (blockIdx.x * blockDim.x + threadIdx.x) >> 5;
    int lane = threadIdx.x & 31;
    if (wave >= n_edges) return;
    int r = rows[wave];
    int c = cols[wave];
    float wv = w[wave];
    const float* s = src + (long long)c * DIM;
    float* d = dst + (long long)r * DIM;
    #pragma unroll
    for (int k = lane; k < DIM; k += 32) {
        unsafeAtomicAdd(&d[k], wv * s[k]);
    }
}

// ---------------------------------------------------------------------------
// 4) GEMM Dst = (ReLU?)(A @ W) with V_WMMA_F32_16X16X4_F32.
//    One wave per 16x16 output tile; K loop = 192 steps of K=4.
//    A layout: lane m = lane&15 (row), kb = (lane>>4)*2, VGPR i -> K = kb+i
//    B layout: lane n = lane&15 (col), same K striping.
//    C layout: VGPR i -> M = i + 8*(lane>>4), N = lane&15.
// ---------------------------------------------------------------------------
template <bool RELU>
__global__ void gemm_wmma_kernel(const float* __restrict__ A,
                                 const float* __restrict__ W,
                                 float* __restrict__ Dst) {
    int wave = threadIdx.x >> 5;
    int lane = threadIdx.x & 31;
    int tile = blockIdx.x * WAVES_PER_BLOCK + wave;
    if (tile >= M_TILES * N_TILES) return;
    int tm = tile / N_TILES;
    int tn = tile % N_TILES;

    int mn = lane & 15;            // A row / B col within tile
    int kb = (lane >> 4) << 1;     // K sub-offset: 0 or 2

    const float* arow = A + (long long)(tm * 16 + mn) * DIM;   // contiguous K
    const float* wcol = W + tn * 16 + mn;                      // stride-DIM K

    v8f c = {};
    #pragma unroll 8
    for (int k = 0; k < DIM; k += 4) {
        v2f a, b;
        a.x = arow[k + kb];
        a.y = arow[k + kb + 1];
        b.x = wcol[(long long)(k + kb) * DIM];
        b.y = wcol[(long long)(k + kb + 1) * DIM];
        c = __builtin_amdgcn_wmma_f32_16x16x4_f32(
                /*neg_a=*/false, a, /*neg_b=*/false, b,
                /*c_mod=*/(short)0, c, /*reuse_a=*/false, /*reuse_b=*/false);
    }

    int row_base = tm * 16 + ((lane >> 4) << 3);   // +8 for upper lane half
    int col = tn * 16 + mn;
    #pragma unroll
    for (int i = 0; i < 8; ++i) {
        float v = c[i];
        if (RELU) v = v > 0.0f ? v : 0.0f;
        Dst[(long long)(row_base + i) * DIM + col] = v;
    }
}

// ---------------------------------------------------------------------------
// 5) Residual scatter-add into EOT slots of the output copy (no collisions:
//    (eot_idx[b], b) unique per b).
// ---------------------------------------------------------------------------
__global__ void scatter_add_kernel(const float* __restrict__ h,
                                   const int* __restrict__ eot_idx,
                                   float* __restrict__ out) {
    long long i = (long long)blockIdx.x * blockDim.x + threadIdx.x;
    long long total = (long long)N_NODES * DIM;
    if (i >= total) return;
    int b = (int)(i / DIM);
    int d = (int)(i % DIM);
    int t = eot_idx[b];
    out[((long long)t * N_NODES + b) * DIM + d] += h[(long long)b * DIM + d];
}

// ---------------------------------------------------------------------------
extern "C" void kernel_launch(void* const* d_in, const int* in_sizes, int n_in,
                              void* d_out, int out_size, void* d_ws, size_t ws_size,
                              hipStream_t stream) {
    const float* emb  = (const float*)d_in[0];
    const int*   eot  = (const int*)d_in[1];
    const int*   erow = (const int*)d_in[2];
    const int*   ecol = (const int*)d_in[3];
    const float* ew   = (const float*)d_in[4];
    const float* W1   = (const float*)d_in[5];
    const float* W2   = (const float*)d_in[6];
    const int n_edges = in_sizes[2];

    float* out  = (float*)d_out;
    float* buf0 = (float*)d_ws;                       // [N_PAD, DIM]
    float* buf1 = buf0 + (long long)N_PAD * DIM;      // [N_PAD, DIM]
    int*   idx_tail = (int*)(out + (long long)L_CTX * N_NODES * DIM);

    const int BLK = 256;
    long long emb4 = (long long)L_CTX * N_NODES * DIM / 4;
    long long pad4 = (long long)N_PAD * DIM / 4;
    long long padElems = (long long)N_PAD * DIM;
    long long nElems   = (long long)N_NODES * DIM;

    // 1) big copy (dominant, bandwidth-bound)
    copy_emb_kernel<<<(int)((emb4 + BLK - 1) / BLK), BLK, 0, stream>>>(
        (const float4*)emb, (float4*)out, emb4);

    // 2) gather EOT rows -> buf0 (padded), write idx tail
    gather_eot_kernel<<<(int)((padElems + BLK - 1) / BLK), BLK, 0, stream>>>(
        emb, eot, buf0, idx_tail);

    // 3) h1 = A @ x0 : buf0 -> buf1
    zero_kernel<<<(int)((pad4 + BLK - 1) / BLK), BLK, 0, stream>>>((float4*)buf1, pad4);
    spmm_kernel<<<(n_edges + WAVES_PER_BLOCK - 1) / WAVES_PER_BLOCK, BLK, 0, stream>>>(
        erow, ecol, ew, buf0, buf1, n_edges);

    // 4) h2 = relu(h1 @ W1) : buf1 -> buf0   (WMMA fp32)
    gemm_wmma_kernel<true><<<(M_TILES * N_TILES) / WAVES_PER_BLOCK, BLK, 0, stream>>>(
        buf1, W1, buf0);

    // 5) h3 = A @ h2 : buf0 -> buf1
    zero_kernel<<<(int)((pad4 + BLK - 1) / BLK), BLK, 0, stream>>>((float4*)buf1, pad4);
    spmm_kernel<<<(n_edges + WAVES_PER_BLOCK - 1) / WAVES_PER_BLOCK, BLK, 0, stream>>>(
        erow, ecol, ew, buf0, buf1, n_edges);

    // 6) h4 = h3 @ W2 : buf1 -> buf0   (WMMA fp32)
    gemm_wmma_kernel<false><<<(M_TILES * N_TILES) / WAVES_PER_BLOCK, BLK, 0, stream>>>(
        buf1, W2, buf0);

    // 7) residual scatter-add into output copy
    scatter_add_kernel<<<(int)((nElems + BLK - 1) / BLK), BLK, 0, stream>>>(
        buf0, eot, out);
}